// OT_Loss_74259984547854
// MI455X (gfx1250) — compile-verified
//
#include <hip/hip_runtime.h>

// Problem constants (must match reference)
#define C_SIZE   512
#define DSF      8
#define W        64
#define S        4096           // W*W
#define N_PTS    1024
#define BATCH    32
#define REGP     10.0f
#define N_ITER   100

#define THREADS     512
#define PER_THREAD  (S / THREADS)   // 8 elements per thread
#define NWAVES      (THREADS / 32)  // 16 wave32s

typedef float v2f __attribute__((ext_vector_type(2)));
typedef float v8f __attribute__((ext_vector_type(8)));

// ---------------------------------------------------------------------------
// All-WMMA exact 32-lane sum (no DS ops, no shuffles).
// WMMA #1: A(16x4) VGPR0 = lane partials, VGPR1 = 0; B = ones(4x16)
//          => D[m][n] = p_m + p_{m+16}.  Per-lane sum of the 8 D VGPRs puts
//          S_lo in every low-half lane, S_hi in every high-half lane.
// WMMA #2 on that value => every D2 element = S_lo + S_hi = total.
// Requires EXEC all-ones (uniform control flow at all call sites).
// ---------------------------------------------------------------------------
__device__ __forceinline__ float wave_sum_wmma(float v) {
    v2f ones; ones[0] = 1.0f; ones[1] = 1.0f;
    v8f c = {};
    v2f a; a[0] = v; a[1] = 0.0f;
    v8f d = __builtin_amdgcn_wmma_f32_16x16x4_f32(
        false, a, false, ones, (short)0, c, false, false);
    float s = ((d[0] + d[1]) + (d[2] + d[3])) + ((d[4] + d[5]) + (d[6] + d[7]));
    v2f a2; a2[0] = s; a2[1] = 0.0f;
    v8f d2 = __builtin_amdgcn_wmma_f32_16x16x4_f32(
        false, a2, false, ones, (short)0, c, false, false);
    return d2[0];
}

// ---------------------------------------------------------------------------
// Wave-level max butterfly: pure VALU via v_permlane16/v_permlanex16 when
// available; __shfl_xor fallback otherwise.
// ---------------------------------------------------------------------------
#if defined(__has_builtin)
#if __has_builtin(__builtin_amdgcn_permlane16) && __has_builtin(__builtin_amdgcn_permlanex16)
#define OT_USE_PERMLANE 1
#endif
#endif

#ifdef OT_USE_PERMLANE
__device__ __forceinline__ float ot_xor16(float x) {
    int i = __float_as_int(x);
    int r = __builtin_amdgcn_permlanex16(i, i, (int)0x76543210, (int)0xFEDCBA98,
                                         false, false);
    return __int_as_float(r);
}
__device__ __forceinline__ float ot_rowperm(float x, int sel_lo, int sel_hi) {
    int i = __float_as_int(x);
    int r = __builtin_amdgcn_permlane16(i, i, sel_lo, sel_hi, false, false);
    return __int_as_float(r);
}
__device__ __forceinline__ float wave_max(float m) {
    m = fmaxf(m, ot_xor16(m));
    m = fmaxf(m, ot_rowperm(m, (int)0xFEDCBA98, (int)0x76543210)); // xor 8
    m = fmaxf(m, ot_rowperm(m, (int)0x32107654, (int)0xBA98FEDC)); // xor 4
    m = fmaxf(m, ot_rowperm(m, (int)0x54761032, (int)0xDCFE98BA)); // xor 2
    m = fmaxf(m, ot_rowperm(m, (int)0x67452301, (int)0xEFCDAB89)); // xor 1
    return m;
}
#else
__device__ __forceinline__ float wave_max(float m) {
#pragma unroll
    for (int off = 16; off > 0; off >>= 1)
        m = fmaxf(m, __shfl_xor(m, off, 32));
    return m;
}
#endif

// Fixed-order vectorized combine of the 16 wave partials (4x ds_load_b128 +
// tree). Deterministic: identical order every call.
__device__ __forceinline__ float lds_sum16(const float* r) {
    const float4* r4 = (const float4*)r;
    float4 a = r4[0], b = r4[1], c = r4[2], d = r4[3];
    float x0 = a.x + b.x, x1 = a.y + b.y, x2 = a.z + b.z, x3 = a.w + b.w;
    float y0 = c.x + d.x, y1 = c.y + d.y, y2 = c.z + d.z, y3 = c.w + d.w;
    return ((x0 + y0) + (x1 + y1)) + ((x2 + y2) + (x3 + y3));
}
__device__ __forceinline__ float lds_max16(const float* r) {
    const float4* r4 = (const float4*)r;
    float4 a = r4[0], b = r4[1], c = r4[2], d = r4[3];
    float x0 = fmaxf(a.x, b.x), x1 = fmaxf(a.y, b.y);
    float x2 = fmaxf(a.z, b.z), x3 = fmaxf(a.w, b.w);
    float y0 = fmaxf(c.x, d.x), y1 = fmaxf(c.y, d.y);
    float y2 = fmaxf(c.z, d.z), y3 = fmaxf(c.w, d.w);
    return fmaxf(fmaxf(fmaxf(x0, y0), fmaxf(x1, y1)),
                 fmaxf(fmaxf(x2, y2), fmaxf(x3, y3)));
}

// ---------------------------------------------------------------------------
// One workgroup per batch. 100 sequential softmax-ASGD steps, all state in
// registers (scaled by 1/REG to save per-element multiplies), cost row folded
// into 2 FMAs per element via softmax shift invariance, all 100 sample points
// pre-gathered into LDS, double-buffered single-barrier reductions.
// ---------------------------------------------------------------------------
__global__ __launch_bounds__(THREADS)
void ot_asgd_kernel(const float* __restrict__ nd,    // normed density  (B, S)
                    const float* __restrict__ ud,    // unnormed density(B, S)
                    const float* __restrict__ pts,   // points (B, N_PTS, 2)
                    const int*   __restrict__ sidx,  // sample_idx (N_ITER, B)
                    float* __restrict__ partial)     // per-batch loss (B)
{
    __shared__ __align__(16) float redsum[2][NWAVES];
    __shared__ __align__(16) float redmax[2][NWAVES];
    __shared__ float s_px[N_ITER];
    __shared__ float s_py[N_ITER];

    const int b    = blockIdx.x;
    const int t    = threadIdx.x;
    const int wave = t >> 5;
    const int lane = t & 31;

    // gfx1250 prefetch of the streamed inputs (global_prefetch_b8)
    __builtin_prefetch(nd + (size_t)b * S, 0, 0);
    __builtin_prefetch(ud + (size_t)b * S, 0, 0);

    // Pre-gather all 100 sample points for this batch into LDS: removes the
    // data-dependent global load from the critical 100-step chain.
    if (t < N_ITER) {
        const int i = sidx[t * BATCH + b];
        const float2 p = ((const float2*)(pts + (size_t)b * N_PTS * 2))[i];
        s_px[t] = p.x;
        s_py[t] = p.y;
    }

    // Per-thread element set: j = t + q*THREADS (coalesced).
    // State kept scaled by 1/REG:  curz = cur/10, avez = ave/10.
    // Aj = log(max(b,1e-38)) - (cr^2 + cc^2)/10   (static per element)
    // z_j = curz_j + fma(0.2*py, cr_j, fma(0.2*px, cc_j, Aj))
    //       (uniform -(px^2+py^2)/10 shift dropped: softmax-invariant)
    float bv[PER_THREAD], Aj[PER_THREAD];
    float cr[PER_THREAD], cc[PER_THREAD];
    float curz[PER_THREAD], avez[PER_THREAD];
#pragma unroll
    for (int q = 0; q < PER_THREAD; ++q) {
        const int j = t + q * THREADS;
        const float bj = nd[(size_t)b * S + j];
        bv[q] = bj;
        const int r = j >> 6, c = j & 63;
        const float crv = (float)(r * DSF + DSF / 2) * (1.0f / 256.0f) - 1.0f;
        const float ccv = (float)(c * DSF + DSF / 2) * (1.0f / 256.0f) - 1.0f;
        cr[q] = crv;
        cc[q] = ccv;
        Aj[q] = __logf(fmaxf(bj, 1e-38f)) - (crv * crv + ccv * ccv) * (1.0f / REGP);
        curz[q] = 0.0f;
        avez[q] = 0.0f;
    }
    __syncthreads();

    const float lr10 = (REGP * (float)N_PTS) * (1.0f / REGP);   // 1024
    int ph = 0;

    for (int k = 1; k <= N_ITER; ++k) {
        const float px2 = (2.0f / REGP) * s_px[k - 1];
        const float py2 = (2.0f / REGP) * s_py[k - 1];

        float z[PER_THREAD];
#pragma unroll
        for (int q = 0; q < PER_THREAD; ++q)
            z[q] = curz[q] + __builtin_fmaf(py2, cr[q],
                             __builtin_fmaf(px2, cc[q], Aj[q]));

        // local 8-max as max3-friendly triple tree
        const float mA = fmaxf(fmaxf(z[0], z[1]), z[2]);
        const float mB = fmaxf(fmaxf(z[3], z[4]), z[5]);
        const float mC = fmaxf(fmaxf(z[6], z[7]), mA);
        float m = fmaxf(mB, mC);

        // --- reduction round 1: block max (1 barrier, double-buffered) ---
        m = wave_max(m);
        if (lane == 0) redmax[ph][wave] = m;
        __syncthreads();
        const float zmax = lds_max16(redmax[ph]);

        float e[PER_THREAD];
        float psum = 0.0f;
#pragma unroll
        for (int q = 0; q < PER_THREAD; ++q) {
            e[q] = __expf(z[q] - zmax);
            psum += e[q];
        }
        // --- reduction round 2: block sum (WMMA wave sum, 1 barrier) ---
        const float ws = wave_sum_wmma(psum);
        if (lane == 0) redsum[ph][wave] = ws;
        __syncthreads();
        const float tot = lds_sum16(redsum[ph]);

        const float inv    = 1.0f / tot;
        const float coef10 = lr10 * rsqrtf((float)k);
        const float invk   = 1.0f / (float)k;
#pragma unroll
        for (int q = 0; q < PER_THREAD; ++q) {
            const float tq = __builtin_fmaf(-inv, e[q], bv[q]);   // b - khi
            curz[q] = __builtin_fmaf(coef10, tq, curz[q]);
            avez[q] = __builtin_fmaf(invk, curz[q] - avez[q], avez[q]);
        }
        ph ^= 1;
    }

    // ---- final loss terms for this batch (beta = REG * avez) ----
    float u[PER_THREAD];
    float psc = 0.0f, pta = 0.0f;
#pragma unroll
    for (int q = 0; q < PER_THREAD; ++q) {
        const int j = t + q * THREADS;
        u[q] = ud[(size_t)b * S + j];
        psc += u[q];
        pta += u[q] * avez[q];
    }
    const float s1 = wave_sum_wmma(psc);
    const float s2 = wave_sum_wmma(pta);
    if (lane == 0) { redsum[ph][wave] = s1; redmax[ph][wave] = s2; }
    __syncthreads();
    const float sc    = lds_sum16(redsum[ph]);
    const float tsum  = REGP * lds_sum16(redmax[ph]);   // fold the x10 here
    const float denom = sc * sc + 1e-8f;
    const float g1_10 = REGP * (sc / denom);            // scale for avez
    const float g2    = tsum / denom;

    float pl = 0.0f;
#pragma unroll
    for (int q = 0; q < PER_THREAD; ++q)
        pl += u[q] * __builtin_fmaf(g1_10, avez[q], -g2);

    const float s3 = wave_sum_wmma(pl);
    ph ^= 1;
    if (lane == 0) redsum[ph][wave] = s3;
    __syncthreads();
    const float lb = lds_sum16(redsum[ph]);
    if (t == 0) partial[b] = lb;
}

// Deterministic fixed-order final accumulation of the 32 batch partials.
__global__ void ot_finalize_kernel(const float* __restrict__ partial,
                                   float* __restrict__ out)
{
    if (threadIdx.x == 0 && blockIdx.x == 0) {
        float s = 0.0f;
        for (int i = 0; i < BATCH; ++i) s += partial[i];
        out[0] = s;
    }
}

extern "C" void kernel_launch(void* const* d_in, const int* in_sizes, int n_in,
                              void* d_out, int out_size, void* d_ws, size_t ws_size,
                              hipStream_t stream)
{
    (void)in_sizes; (void)n_in; (void)out_size; (void)ws_size;
    const float* nd   = (const float*)d_in[0];   // normed_density  (B,1,W,W)
    const float* ud   = (const float*)d_in[1];   // unnormed_density(B,1,W,W)
    const float* pts  = (const float*)d_in[2];   // points (B,N_PTS,2)
    const int*   sidx = (const int*)d_in[3];     // sample_idx (N_ITER,B)
    float* out     = (float*)d_out;
    float* partial = (float*)d_ws;               // 32 floats of scratch

    ot_asgd_kernel<<<BATCH, THREADS, 0, stream>>>(nd, ud, pts, sidx, partial);
    ot_finalize_kernel<<<1, 32, 0, stream>>>(partial, out);
}